// IncrementalSelfAttention_21715354649641
// MI455X (gfx1250) — compile-verified
//
#include <hip/hip_runtime.h>
#include <hip/hip_bf16.h>

// Problem constants
#define BB 32
#define DD 1024
#define HH 16
#define HD 64
#define SPAST 4096
#define STOT 4097
#define NSPLIT 8
#define SLEN 513            // ceil(4097/8)
#define PART_STRIDE 1032    // m, l, pad..., u[1024] at +8

// d_out layout (floats): attn_output[32*1024] | comb_key[32*4097*1024] | comb_value[...]
#define OUT_K_OFF 32768
#define OUT_V_OFF (32768 + 32*4097*1024)          // 134283264

// d_ws layout (floats)
#define Q_OFF    0                                 // 32768
#define QW_OFF   32768                             // 32*16*1024 = 524288
#define QB_OFF   (32768 + 524288)                  // 512
#define CTX_OFF  (QB_OFF + 512)                    // 32768
#define PART_OFF (CTX_OFF + 32768)                 // 32*16*8*1032 = 4227072
// total ~4.82M floats = ~19.3 MB workspace

typedef float v2f __attribute__((ext_vector_type(2)));
typedef float v8f __attribute__((ext_vector_type(8)));
typedef int vint4 __attribute__((vector_size(16)));

// gfx1250 async global<->LDS data movers (ASYNCcnt-tracked), if the toolchain
// exposes them; otherwise fall back to the synchronous VGPR path.
#if __has_builtin(__builtin_amdgcn_global_load_async_to_lds_b128) && \
    __has_builtin(__builtin_amdgcn_global_store_async_from_lds_b128) && \
    __has_builtin(__builtin_amdgcn_s_wait_asynccnt)
#define HAVE_ASYNC_LDS 1
#else
#define HAVE_ASYNC_LDS 0
#endif

#define AS1 __attribute__((address_space(1)))
#define AS3 __attribute__((address_space(3)))

__device__ inline float waveSum32(float v) {
#pragma unroll
  for (int off = 16; off > 0; off >>= 1)
    v += __shfl_xor(v, off, 32);
  return v;
}

// ---------------------------------------------------------------------------
// f32 WMMA GEMM: Y(32x1024) = X(32x1024) @ W(1024x1024, row=[n][k]) ^T + bias
// One wave per 16x16 output tile; K loop in steps of 4 (V_WMMA_F32_16X16X4_F32).
// A layout: lane L: m=L&15, khalf=L>>4 holds A[m][k0+2*kh], A[m][k0+2*kh+1]
// B layout (mirrored): lane L: n=L&15 holds B[k0+2*kh][n], B[k0+2*kh+1][n]
// C/D layout: VGPR r: m = r + 8*(L>=16), n = L&15
// ---------------------------------------------------------------------------
__global__ __launch_bounds__(32) void gemm32_f32wmma(
    const float* __restrict__ X, const float* __restrict__ W,
    const float* __restrict__ bias, float* __restrict__ Y) {
  const int lane = threadIdx.x;
  const int nt = blockIdx.x, mt = blockIdx.y;
  const int half = lane >> 4, lo = lane & 15;
  const float* arow = X + (size_t)(mt * 16 + lo) * DD;
  const float* brow = W + (size_t)(nt * 16 + lo) * DD;
  v8f c = {0.f, 0.f, 0.f, 0.f, 0.f, 0.f, 0.f, 0.f};
  for (int k0 = 0; k0 < DD; k0 += 4) {
    v2f a, bb;
    a.x = arow[k0 + 2 * half];
    a.y = arow[k0 + 2 * half + 1];
    bb.x = brow[k0 + 2 * half];
    bb.y = brow[k0 + 2 * half + 1];
    c = __builtin_amdgcn_wmma_f32_16x16x4_f32(false, a, false, bb, (short)0, c,
                                              false, false);
  }
  const float bv = bias[nt * 16 + lo];
#pragma unroll
  for (int r = 0; r < 8; ++r)
    Y[(size_t)(mt * 16 + r + 8 * half) * DD + nt * 16 + lo] = c[r] + bv;
}

// ---------------------------------------------------------------------------
// qW[b,h,c] = 0.125 * sum_d q[b, h*64+d] * Wk[h*64+d, c]   (Wk = in_proj_w+1024 rows)
// Per h: GEMM M=32(b) N=1024(c) K=64, f32 WMMA.
// ---------------------------------------------------------------------------
__global__ __launch_bounds__(32) void qw_f32wmma(
    const float* __restrict__ q, const float* __restrict__ Wfull,
    float* __restrict__ qW) {
  const int lane = threadIdx.x;
  const int nt = blockIdx.x, mt = blockIdx.y, h = blockIdx.z;
  const int half = lane >> 4, lo = lane & 15;
  const float* arow = q + (size_t)(mt * 16 + lo) * DD + h * HD;
  const float* bcol = Wfull + (size_t)(DD + h * HD) * DD + nt * 16 + lo;
  v8f c = {0.f, 0.f, 0.f, 0.f, 0.f, 0.f, 0.f, 0.f};
  for (int k0 = 0; k0 < HD; k0 += 4) {
    v2f a, bb;
    a.x = arow[k0 + 2 * half];
    a.y = arow[k0 + 2 * half + 1];
    bb.x = bcol[(size_t)(k0 + 2 * half) * DD];
    bb.y = bcol[(size_t)(k0 + 2 * half + 1) * DD];
    c = __builtin_amdgcn_wmma_f32_16x16x4_f32(false, a, false, bb, (short)0, c,
                                              false, false);
  }
#pragma unroll
  for (int r = 0; r < 8; ++r) {
    const int bidx = mt * 16 + r + 8 * half;  // batch index
    qW[((size_t)bidx * HH + h) * DD + nt * 16 + lo] = c[r] * 0.125f;
  }
}

// qb[b,h] = 0.125 * sum_d q[b,h*64+d] * bk[h*64+d]
__global__ void qb_kernel(const float* __restrict__ q,
                          const float* __restrict__ bvec,
                          float* __restrict__ qb) {
  const int i = blockIdx.x * blockDim.x + threadIdx.x;
  if (i >= BB * HH) return;
  const int b = i >> 4, h = i & 15;
  float acc = 0.f;
  for (int d = 0; d < HD; ++d)
    acc += q[(size_t)b * DD + h * HD + d] * bvec[DD + h * HD + d];
  qb[i] = acc * 0.125f;
}

// ---------------------------------------------------------------------------
// Main fused streaming kernel. grid = (32 batches, 8 S-splits), 256 threads.
// Per group of 8 rows:
//   1) async-load past_k/v rows straight into LDS (bypasses VGPRs)
//   2) s_wait_asynccnt 0 + barrier
//   3) fire async LDS->global stores for the comb_key/comb_value copy
//      (overlaps with compute)
//   4) each wave (2 heads) computes scores x.qW+qb with online softmax and
//      accumulates u[h] += e * x_v entirely in registers
//   5) s_wait_asynccnt 0 + barrier before LDS buffer reuse
// Partials {m, l, u[1024]} per (b,h,split) -> ws.
// ---------------------------------------------------------------------------
__global__ __launch_bounds__(256) void attn_stream(
    const float* __restrict__ key, const float* __restrict__ value,
    const float* __restrict__ past_key, const float* __restrict__ past_value,
    const float* __restrict__ qW, const float* __restrict__ qb,
    float* __restrict__ out, float* __restrict__ part) {
  __shared__ float s_xk[8][DD];
  __shared__ float s_xv[8][DD];
  const int b = blockIdx.x, split = blockIdx.y;
  const int tid = threadIdx.x, lane = tid & 31, wave = tid >> 5;
  const int h0 = 2 * wave, h1 = h0 + 1;
  const int rowBeg = split * SLEN;
  const int rowEnd = min(rowBeg + SLEN, STOT);

  float qw0[32], qw1[32], u0[32], u1[32];
  {
    const float* g0 = qW + ((size_t)(b * HH + h0)) * DD + lane;
    const float* g1 = qW + ((size_t)(b * HH + h1)) * DD + lane;
#pragma unroll
    for (int j = 0; j < 32; ++j) {
      qw0[j] = g0[32 * j];
      qw1[j] = g1[32 * j];
      u0[j] = 0.f;
      u1[j] = 0.f;
    }
  }
  const float qb0 = qb[b * HH + h0], qb1 = qb[b * HH + h1];
  float m0 = -3.0e38f, m1 = -3.0e38f, l0 = 0.f, l1 = 0.f;

  float* combK = out + OUT_K_OFF + (size_t)b * STOT * DD;
  float* combV = out + OUT_V_OFF + (size_t)b * STOT * DD;

  for (int g = rowBeg; g < rowEnd; g += 8) {
    const int n = min(8, rowEnd - g);
    // ---- stage rows into LDS ----
    for (int idx = tid; idx < n * 256; idx += 256) {
      const int r = idx >> 8;
      const int c4 = (idx & 255) << 2;
      const int s = g + r;
      const float* srcK = (s < SPAST)
                              ? past_key + ((size_t)b * SPAST + s) * DD + c4
                              : key + (size_t)b * DD + c4;
      const float* srcV = (s < SPAST)
                              ? past_value + ((size_t)b * SPAST + s) * DD + c4
                              : value + (size_t)b * DD + c4;
#if HAVE_ASYNC_LDS
      __builtin_amdgcn_global_load_async_to_lds_b128(
          (AS1 vint4*)(void*)srcK, (AS3 vint4*)&s_xk[r][c4], 0, 0);
      __builtin_amdgcn_global_load_async_to_lds_b128(
          (AS1 vint4*)(void*)srcV, (AS3 vint4*)&s_xv[r][c4], 0, 0);
#else
      const float4 fk = *(const float4*)srcK;
      const float4 fv = *(const float4*)srcV;
      *(float4*)(combK + (size_t)s * DD + c4) = fk;
      *(float4*)(combV + (size_t)s * DD + c4) = fv;
      *(float4*)(&s_xk[r][c4]) = fk;
      *(float4*)(&s_xv[r][c4]) = fv;
#endif
    }
#if HAVE_ASYNC_LDS
    __builtin_amdgcn_s_wait_asynccnt(0);
#endif
    __syncthreads();
#if HAVE_ASYNC_LDS
    // ---- fire the comb copy LDS->global; completes during compute ----
    for (int idx = tid; idx < n * 256; idx += 256) {
      const int r = idx >> 8;
      const int c4 = (idx & 255) << 2;
      const int s = g + r;
      __builtin_amdgcn_global_store_async_from_lds_b128(
          (AS1 vint4*)(combK + (size_t)s * DD + c4), (AS3 vint4*)&s_xk[r][c4],
          0, 0);
      __builtin_amdgcn_global_store_async_from_lds_b128(
          (AS1 vint4*)(combV + (size_t)s * DD + c4), (AS3 vint4*)&s_xv[r][c4],
          0, 0);
    }
#endif
    // ---- compute ----
    for (int r = 0; r < n; ++r) {
      float a0 = 0.f, a1 = 0.f;
#pragma unroll
      for (int j = 0; j < 32; ++j) {
        const float x = s_xk[r][lane + 32 * j];
        a0 += x * qw0[j];
        a1 += x * qw1[j];
      }
      const float sc0 = waveSum32(a0) + qb0;
      const float sc1 = waveSum32(a1) + qb1;
      float e0, e1;
      if (sc0 > m0) {  // wave-uniform branch; rescale is rare
        const float f = __expf(m0 - sc0);
        l0 = l0 * f + 1.f;
#pragma unroll
        for (int j = 0; j < 32; ++j) u0[j] *= f;
        m0 = sc0;
        e0 = 1.f;
      } else {
        e0 = __expf(sc0 - m0);
        l0 += e0;
      }
      if (sc1 > m1) {
        const float f = __expf(m1 - sc1);
        l1 = l1 * f + 1.f;
#pragma unroll
        for (int j = 0; j < 32; ++j) u1[j] *= f;
        m1 = sc1;
        e1 = 1.f;
      } else {
        e1 = __expf(sc1 - m1);
        l1 += e1;
      }
#pragma unroll
      for (int j = 0; j < 32; ++j) {
        const float xv = s_xv[r][lane + 32 * j];
        u0[j] += e0 * xv;
        u1[j] += e1 * xv;
      }
    }
#if HAVE_ASYNC_LDS
    __builtin_amdgcn_s_wait_asynccnt(0);  // comb stores done before LDS reuse
#endif
    __syncthreads();
  }
  // write split partials: {m, l, u[1024]}
  float* pb0 = part + (((size_t)(b * HH + h0)) * NSPLIT + split) * PART_STRIDE;
  float* pb1 = part + (((size_t)(b * HH + h1)) * NSPLIT + split) * PART_STRIDE;
  if (lane == 0) {
    pb0[0] = m0; pb0[1] = l0;
    pb1[0] = m1; pb1[1] = l1;
  }
#pragma unroll
  for (int j = 0; j < 32; ++j) {
    pb0[8 + lane + 32 * j] = u0[j];
    pb1[8 + lane + 32 * j] = u1[j];
  }
}

// ---------------------------------------------------------------------------
// Combine splits + V head projection:
// ctx[b, h*64+d] = Wv[h*64+d,:] . (sum_i e_i u_i / L) + bv[h*64+d]
// grid = (16 heads, 32 batches), 64 threads (one output d per thread).
// ---------------------------------------------------------------------------
__global__ __launch_bounds__(64) void combine_ctx(
    const float* __restrict__ part, const float* __restrict__ Wfull,
    const float* __restrict__ bvec, float* __restrict__ ctx) {
  __shared__ float ubar[DD];
  const int h = blockIdx.x, b = blockIdx.y;
  const int tid = threadIdx.x;
  const float* pb = part + ((size_t)(b * HH + h)) * NSPLIT * PART_STRIDE;
  float m[NSPLIT], l[NSPLIT], e[NSPLIT];
  float M = -3.0e38f;
#pragma unroll
  for (int i = 0; i < NSPLIT; ++i) {
    m[i] = pb[i * PART_STRIDE];
    l[i] = pb[i * PART_STRIDE + 1];
    M = fmaxf(M, m[i]);
  }
  float L = 0.f;
#pragma unroll
  for (int i = 0; i < NSPLIT; ++i) {
    e[i] = __expf(m[i] - M);
    L += l[i] * e[i];
  }
  const float invL = 1.f / L;
  for (int c = tid; c < DD; c += 64) {
    float s = 0.f;
#pragma unroll
    for (int i = 0; i < NSPLIT; ++i) s += pb[i * PART_STRIDE + 8 + c] * e[i];
    ubar[c] = s * invL;
  }
  __syncthreads();
  const int d = tid;
  const float* wrow = Wfull + (size_t)(2 * DD + h * HD + d) * DD;  // Wv row
  float acc = 0.f;
  for (int c = 0; c < DD; ++c) acc += wrow[c] * ubar[c];
  ctx[(size_t)b * DD + h * HD + d] = acc + bvec[2 * DD + h * HD + d];
}

extern "C" void kernel_launch(void* const* d_in, const int* in_sizes, int n_in,
                              void* d_out, int out_size, void* d_ws,
                              size_t ws_size, hipStream_t stream) {
  const float* query      = (const float*)d_in[0];
  const float* key        = (const float*)d_in[1];
  const float* value      = (const float*)d_in[2];
  const float* past_key   = (const float*)d_in[3];
  const float* past_value = (const float*)d_in[4];
  const float* in_proj_w  = (const float*)d_in[5];
  const float* in_proj_b  = (const float*)d_in[6];
  const float* out_proj_w = (const float*)d_in[7];
  const float* out_proj_b = (const float*)d_in[8];
  float* out = (float*)d_out;
  float* ws = (float*)d_ws;

  float* q_ws   = ws + Q_OFF;
  float* qW_ws  = ws + QW_OFF;
  float* qb_ws  = ws + QB_OFF;
  float* ctx_ws = ws + CTX_OFF;
  float* part   = ws + PART_OFF;

  // 1) q = query @ Wq^T + bq   (f32 WMMA)
  gemm32_f32wmma<<<dim3(64, 2), 32, 0, stream>>>(query, in_proj_w, in_proj_b,
                                                 q_ws);
  // 2) qW[b,h,:] = scale * q_head @ Wk_head ; qb[b,h] = scale * q_head.bk_head
  qw_f32wmma<<<dim3(64, 2, 16), 32, 0, stream>>>(q_ws, in_proj_w, qW_ws);
  qb_kernel<<<2, 256, 0, stream>>>(q_ws, in_proj_b, qb_ws);
  // 3) fused cache-concat copy + streaming split-softmax attention
  attn_stream<<<dim3(32, NSPLIT), 256, 0, stream>>>(
      key, value, past_key, past_value, qW_ws, qb_ws, out, part);
  // 4) combine splits + V projection per head -> ctx
  combine_ctx<<<dim3(16, 32), 64, 0, stream>>>(part, in_proj_w, in_proj_b,
                                               ctx_ws);
  // 5) attn_output = ctx @ Wo^T + bo   (f32 WMMA)
  gemm32_f32wmma<<<dim3(64, 2), 32, 0, stream>>>(ctx_ws, out_proj_w,
                                                 out_proj_b, out);
}